// Advect_63702954934311
// MI455X (gfx1250) — compile-verified
//
#include <hip/hip_runtime.h>

// Advection with minmod slope limiter along axis 1 of (16, 4100, 1024) fp32.
// Memory-bound stencil: register rolling-window along j, float4 (B128)
// coalesced access along i, NT stores, CDNA5 global_prefetch.
// v2: batched prologue loads (single wait), early load issue in the loop,
//     2x unroll for ILP / loop-overhead amortization.

#define THETA 2.0f

constexpr int NB  = 16;     // batch
constexpr int NJ  = 4100;   // input rows (stencil axis)
constexpr int NJO = 4096;   // output rows
constexpr int NI  = 1024;   // contiguous axis
constexpr int CJ  = 64;     // output rows per block (j-chunk)

typedef __attribute__((ext_vector_type(4))) float f32x4;

__device__ __forceinline__ float minmod3(float a, float b, float c) {
    float s0 = THETA * (b - a);
    float s1 = 0.5f  * (c - a);
    float s2 = THETA * (c - b);
    float mn = fminf(fminf(s0, s1), s2);
    float mx = fmaxf(fmaxf(s0, s1), s2);
    return (mn < 0.0f) ? fminf(mx, 0.0f) : mn;
}

__global__ __launch_bounds__(256) void advect_minmod_kernel(
    const float* __restrict__ rho,
    const float* __restrict__ v,
    float* __restrict__ out)
{
    const int i  = threadIdx.x * 4;          // 0..1020, 16B-aligned column group
    const int b  = blockIdx.x;               // batch
    const int j0 = blockIdx.y * CJ;          // first output row of this block

    const float* rbase = rho + (size_t)b * NJ  * NI + i;
    const float* vbase = v   + (size_t)b * NJ  * NI + i;
    float*       obase = out + (size_t)b * NJO * NI + i;

    // ---- Prologue: issue all 10 loads, single wait, then compute ----
    f32x4 rld[5], wld[5];
    #pragma unroll
    for (int k = 0; k < 5; ++k) {
        rld[k] = *(const f32x4*)(rbase + (size_t)(j0 + k) * NI);
        wld[k] = *(const f32x4*)(vbase + (size_t)(j0 + k) * NI);
    }

    // Rolling windows: F[k] = rho*v at row j+k ; W[k] = v at row j+k.
    f32x4 F[5], W[5], S[3], fn0, fn1;
    #pragma unroll
    for (int k = 0; k < 5; ++k) {
        W[k] = wld[k];
        #pragma unroll
        for (int c = 0; c < 4; ++c) F[k][c] = rld[k][c] * wld[k][c];
    }

    // S[0..2] = slopes at rows j0+1 .. j0+3
    #pragma unroll
    for (int c = 0; c < 4; ++c) {
        S[0][c] = minmod3(F[0][c], F[1][c], F[2][c]);
        S[1][c] = minmod3(F[1][c], F[2][c], F[3][c]);
        S[2][c] = minmod3(F[2][c], F[3][c], F[4][c]);
    }

    // fn0 = flux_net at m = j0
    {
        const bool mZero = (j0 == 0);   // flux_plus forced to 0 at m==0
        #pragma unroll
        for (int c = 0; c < 4; ++c) {
            float fm = (W[2][c] >= 0.0f) ? 0.0f : (F[2][c] - 0.5f * S[1][c]);
            float fp = (W[1][c] <= 0.0f) ? 0.0f : (F[1][c] + 0.5f * S[0][c]);
            if (mZero) fp = 0.0f;
            fn0[c] = fm + fp;
        }
    }

    #pragma unroll 2
    for (int jj = 0; jj < CJ; ++jj) {
        const int j = j0 + jj;

        // ---- Issue next-row loads FIRST so they overlap compute+store ----
        int r = j + 5; if (r > NJ - 1) r = NJ - 1;   // clamped value never used
        f32x4 rn = *(const f32x4*)(rbase + (size_t)r * NI);
        f32x4 wn = *(const f32x4*)(vbase + (size_t)r * NI);

        // Prefetch 8 rows ahead of the in-flight load (global_prefetch_b8)
        int rp = j + 13; if (rp > NJ - 1) rp = NJ - 1;
        __builtin_prefetch(rbase + (size_t)rp * NI, 0, 1);
        __builtin_prefetch(vbase + (size_t)rp * NI, 0, 1);

        // fn1 = flux_net at m=j+1: uses F[j+2],F[j+3], S[j+2],S[j+3], V[j+2],V[j+3]
        const bool mLast = (j + 1 == NJO);  // flux_minus forced to 0 at m==4096
        #pragma unroll
        for (int c = 0; c < 4; ++c) {
            float fm = (W[3][c] >= 0.0f) ? 0.0f : (F[3][c] - 0.5f * S[2][c]);
            float fp = (W[2][c] <= 0.0f) ? 0.0f : (F[2][c] + 0.5f * S[1][c]);
            if (mLast) fm = 0.0f;
            fn1[c] = fm + fp;
        }

        // out[j] = flux_net[j] - flux_net[j+1]   (streaming write -> NT hint)
        f32x4 o;
        #pragma unroll
        for (int c = 0; c < 4; ++c) o[c] = fn0[c] - fn1[c];
        __builtin_nontemporal_store(o, (f32x4*)(obase + (size_t)j * NI));

        // ---- Consume the loads: shift windows, compute the new slope ----
        #pragma unroll
        for (int k = 0; k < 4; ++k) { F[k] = F[k + 1]; W[k] = W[k + 1]; }
        W[4] = wn;
        #pragma unroll
        for (int c = 0; c < 4; ++c) F[4][c] = rn[c] * wn[c];

        S[0] = S[1]; S[1] = S[2];
        #pragma unroll
        for (int c = 0; c < 4; ++c)
            S[2][c] = minmod3(F[2][c], F[3][c], F[4][c]);

        fn0 = fn1;
    }
}

extern "C" void kernel_launch(void* const* d_in, const int* in_sizes, int n_in,
                              void* d_out, int out_size, void* d_ws, size_t ws_size,
                              hipStream_t stream)
{
    (void)in_sizes; (void)n_in; (void)out_size; (void)d_ws; (void)ws_size;
    const float* rho = (const float*)d_in[0];
    const float* v   = (const float*)d_in[1];
    // d_in[2] is the python scalar `axis` (==1); layout is baked into the kernel.
    float* out = (float*)d_out;

    dim3 grid(NB, NJO / CJ, 1);   // 16 x 64 = 1024 workgroups, 8 wave32 each
    dim3 block(256, 1, 1);        // covers full 1024-wide contiguous axis (x4 vec)
    advect_minmod_kernel<<<grid, block, 0, stream>>>(rho, v, out);
}